// CapsLayer2D_19370302505141
// MI455X (gfx1250) — compile-verified
//
#include <hip/hip_runtime.h>
#include <hip/hip_bf16.h>
#include <math.h>

// Problem constants (from reference):
// in_caps: (B=16, I=32, Din=8, H=20, W=20) fp32
// W:       (I=32, J=10, Dout=16, Din=8, 3, 3) fp32
// u_hat:   (B, I, J, Dout, H, W) = 32,768,000 fp32 (131 MB, fits in 192MB L2)
// v (out): (B, J, Dout, H, W)    = 1,024,000 fp32

#define BB 16
#define II 32
#define JJ 10
#define DD 16    // Dout
#define DIN 8
#define HW 400   // 20*20
#define KTOT 72  // Din*3*3
#define OCH 160  // J*Dout
#define WPAD 74  // padded K-stride for weight LDS (banks distinct, 8B aligned)

typedef __attribute__((ext_vector_type(2))) float v2f;
typedef __attribute__((ext_vector_type(4))) float v4f;
typedef __attribute__((ext_vector_type(8))) float v8f;

// Compile-time im2col LDS offset for flattened K index (din*9 + dy*3 + dx)
// into the zero-padded 22x22 image: din*484 + dy*22 + dx.
__host__ __device__ constexpr int koff(int k) {
    return (k / 9) * 484 + ((k % 9) / 3) * 22 + ((k % 9) % 3);
}

// ---------------------------------------------------------------------------
// Kernel 1: grouped 3x3 conv as im2col-GEMM per (b,i) using fp32 WMMA.
// GEMM: M=400 (spatial), N=160 (J*Dout), K=72 (Din*9).
// One workgroup (256 thr = 8 waves) per (b,i). Input tile staged into LDS with
// a zero border (22x22, no bounds checks); weights staged o-major with pad-74
// stride so B fragments are one aligned 8-byte LDS load with conflict-free
// banking. K-loop fully unrolled -> 180 static v_wmma, no div/mod in the loop.
// ---------------------------------------------------------------------------
__global__ __launch_bounds__(256) void caps_conv_wmma(const float* __restrict__ x,
                                                      const float* __restrict__ w,
                                                      float* __restrict__ u) {
    __shared__ float wsh[OCH * WPAD];     // 47360 B : wsh[o*74 + kk]
    __shared__ float xsh[DIN * 22 * 22];  // 15488 B : zero-padded image

    const int blk = blockIdx.x;      // b*32 + i
    const int i   = blk & 31;
    const int tid = threadIdx.x;

    // Zero the padded image border region (just zero it all: 3872 floats).
    for (int t = tid; t < DIN * 22 * 22; t += 256) xsh[t] = 0.0f;
    __syncthreads();
    // Fill interior.
    const float* xin = x + (size_t)blk * DIN * HW;
    for (int t = tid; t < DIN * HW; t += 256) {
        int din = t / HW, p = t % HW;
        int h = p / 20, wd = p % 20;
        xsh[din * 484 + (h + 1) * 22 + (wd + 1)] = xin[t];
    }
    // Stage weights o-major: wsh[o*74 + kk] = wg[o*72 + kk]  (contiguous src,
    // float2-vectorized: 5760 float2 elements, both sides 8B aligned).
    const float* wg = w + (size_t)i * OCH * KTOT;
    for (int t = tid; t < OCH * (KTOT / 2); t += 256) {
        int o = t / 36, q = t % 36;
        *(v2f*)&wsh[o * WPAD + 2 * q] = *(const v2f*)&wg[o * KTOT + 2 * q];
    }
    __syncthreads();

    const int wave = tid >> 5;
    const int lane = tid & 31;
    const int mrow = lane & 15;            // M row within tile / N col within tile
    const bool up  = (lane >= 16);         // lanes 16-31 hold K+2 / K+3
    const int khalf = up ? 2 : 0;

    // Per-lane B row base: column o = n*16 + mrow, K half-select folded in.
    const float* wp0 = &wsh[mrow * WPAD + khalf];

    for (int mt = wave; mt < 25; mt += 8) {
        const int p0 = mt * 16;
        const int m  = p0 + mrow;          // spatial position of this lane's A row
        const float* xrow = &xsh[(m / 20) * 22 + (m % 20)];

        v8f acc[10] = {};

#pragma unroll
        for (int k4 = 0; k4 < KTOT; k4 += 4) {
            // A fragment (16x4 f32): lane holds (M=mrow, K=k4+khalf) and +1.
            // Offsets are compile-time constants after full unroll.
            v2f a;
            a.x = xrow[up ? koff(k4 + 2) : koff(k4 + 0)];
            a.y = xrow[up ? koff(k4 + 3) : koff(k4 + 1)];
#pragma unroll
            for (int n = 0; n < 10; ++n) {
                // B fragment: rows K=k4+khalf, +1 at column n*16+mrow -> one b64.
                v2f bf = *(const v2f*)(wp0 + n * (16 * WPAD) + k4);
                acc[n] = __builtin_amdgcn_wmma_f32_16x16x4_f32(
                    /*neg_a=*/false, a, /*neg_b=*/false, bf,
                    /*c_mod=*/(short)0, acc[n],
                    /*reuse_a=*/false, /*reuse_b=*/false);
            }
        }

        // Store D: VGPR r -> M = p0 + r (+8 for lanes 16-31), N = lane&15.
        // 8 consecutive floats per lane, 32B aligned -> two b128 stores.
        const int mo = p0 + (up ? 8 : 0);
        float* ub = u + (size_t)blk * OCH * HW;
#pragma unroll
        for (int n = 0; n < 10; ++n) {
            float* dst = ub + (size_t)(n * 16 + mrow) * HW + mo;
            v4f lo = __builtin_shufflevector(acc[n], acc[n], 0, 1, 2, 3);
            v4f hi = __builtin_shufflevector(acc[n], acc[n], 4, 5, 6, 7);
            *(v4f*)dst = lo;
            *(v4f*)(dst + 4) = hi;
        }
    }
}

// ---------------------------------------------------------------------------
// Zero the routing logits b (B*I*J = 5120 floats).
// ---------------------------------------------------------------------------
__global__ void caps_init_b(float* __restrict__ b_arr) {
    int t = blockIdx.x * 256 + threadIdx.x;
    if (t < BB * II * JJ) b_arr[t] = 0.0f;
}

// ---------------------------------------------------------------------------
// c = softmax_i(b). One wave32 per (b, j); lane = i.
// ---------------------------------------------------------------------------
__global__ void caps_softmax(const float* __restrict__ b_arr, float* __restrict__ c_arr) {
    const int blk = blockIdx.x;           // b*10 + j
    const int b = blk / JJ, j = blk % JJ;
    const int i = threadIdx.x;            // 0..31
    const int idx = (b * II + i) * JJ + j;
    float x = b_arr[idx];
    float mx = x;
    for (int off = 16; off > 0; off >>= 1) mx = fmaxf(mx, __shfl_xor(mx, off, 32));
    float e = __expf(x - mx);
    float s = e;
    for (int off = 16; off > 0; off >>= 1) s += __shfl_xor(s, off, 32);
    c_arr[idx] = e / s;
}

// ---------------------------------------------------------------------------
// Fused weighted-sum + squash: v[b,j,:,h,w] = squash(sum_i c[b,i,j]*u_hat[b,i,j,:,h,w])
// One block per (b, j); threads stride over the 400 contiguous spatial positions
// so every global load is coalesced. u_hat stays in L2.
// ---------------------------------------------------------------------------
__global__ __launch_bounds__(256) void caps_sv(const float* __restrict__ c,
                                               const float* __restrict__ u,
                                               float* __restrict__ v) {
    __shared__ float csh[II];
    const int blk = blockIdx.x;           // b*10 + j
    const int b = blk / JJ, j = blk % JJ;
    const int tid = threadIdx.x;
    if (tid < II) csh[tid] = c[(b * II + tid) * JJ + j];
    __syncthreads();

    for (int p = tid; p < HW; p += 256) {
        float s[DD];
#pragma unroll
        for (int d = 0; d < DD; ++d) s[d] = 0.0f;
        for (int i = 0; i < II; ++i) {
            const float ci = csh[i];
            const float* ub = u + ((size_t)((b * II + i) * JJ + j) * DD) * HW + p;
#pragma unroll
            for (int d = 0; d < DD; ++d) s[d] += ci * ub[(size_t)d * HW];
        }
        float n = 0.0f;
#pragma unroll
        for (int d = 0; d < DD; ++d) n += s[d] * s[d];
        const float scale = (n > 0.0f) ? (n / (sqrtf(n) * (1.0f + n))) : 0.0f;
        float* vb = v + ((size_t)(b * JJ + j) * DD) * HW + p;
#pragma unroll
        for (int d = 0; d < DD; ++d) vb[(size_t)d * HW] = s[d] * scale;
    }
}

// ---------------------------------------------------------------------------
// Agreement: b[b,i,j] += sum_{d,h,w} u_hat[b,i,j,d,h,w] * v[b,j,d,h,w]
// One block per (b,i,j); 6400-element dot, float4-vectorized (both operand
// blocks are 16*400 contiguous, 16B aligned); wave shuffle + LDS reduction.
// ---------------------------------------------------------------------------
__global__ __launch_bounds__(256) void caps_agree(const float* __restrict__ u,
                                                  const float* __restrict__ v,
                                                  float* __restrict__ b_arr) {
    __shared__ float red[8];
    const int blk = blockIdx.x;           // (b*32+i)*10 + j
    const int j  = blk % JJ;
    const int bi = blk / JJ;              // b*32 + i
    const int b  = bi >> 5;
    const int tid = threadIdx.x;

    const v4f* ub = (const v4f*)(u + (size_t)blk * DD * HW);
    const v4f* vb = (const v4f*)(v + ((size_t)(b * JJ + j) * DD) * HW);
    float acc = 0.0f;
    for (int t = tid; t < (DD * HW) / 4; t += 256) {
        __builtin_prefetch(ub + t + 512, 0, 0);   // global_prefetch_b8, streaming
        v4f a = ub[t], c = vb[t];
        acc += a.x * c.x + a.y * c.y + a.z * c.z + a.w * c.w;
    }
    for (int off = 16; off > 0; off >>= 1) acc += __shfl_xor(acc, off, 32);
    if ((tid & 31) == 0) red[tid >> 5] = acc;
    __syncthreads();
    if (tid == 0) {
        float t = 0.0f;
#pragma unroll
        for (int k = 0; k < 8; ++k) t += red[k];
        b_arr[blk] += t;
    }
}

// ---------------------------------------------------------------------------
// Launch: conv -> [softmax -> s/squash -> (agree)] x 3. All on `stream`,
// graph-capture safe, deterministic (b re-zeroed every call).
// Workspace layout: u_hat (32,768,000 f32) | b (5120 f32) | c (5120 f32).
// ---------------------------------------------------------------------------
extern "C" void kernel_launch(void* const* d_in, const int* in_sizes, int n_in,
                              void* d_out, int out_size, void* d_ws, size_t ws_size,
                              hipStream_t stream) {
    (void)in_sizes; (void)n_in; (void)out_size; (void)ws_size;
    const float* in_caps = (const float*)d_in[0];
    const float* Wt      = (const float*)d_in[1];
    float* v_out = (float*)d_out;
    float* u_hat = (float*)d_ws;
    float* b_arr = u_hat + (size_t)BB * II * JJ * DD * HW; // +32,768,000
    float* c_arr = b_arr + BB * II * JJ;                   // +5120

    caps_init_b<<<dim3((BB * II * JJ + 255) / 256), dim3(256), 0, stream>>>(b_arr);
    caps_conv_wmma<<<dim3(BB * II), dim3(256), 0, stream>>>(in_caps, Wt, u_hat);

    for (int it = 0; it < 3; ++it) {
        caps_softmax<<<dim3(BB * JJ), dim3(32), 0, stream>>>(b_arr, c_arr);
        caps_sv<<<dim3(BB * JJ), dim3(256), 0, stream>>>(c_arr, u_hat, v_out);
        if (it < 2) {
            caps_agree<<<dim3(BB * II * JJ), dim3(256), 0, stream>>>(u_hat, v_out, b_arr);
        }
    }
}